// Net_42563125903970
// MI455X (gfx1250) — compile-verified
//
#include <hip/hip_runtime.h>

#define TAGS 64
#define TSTEPS 512
#define BATCH 1024
#define START_TAG 62
#define STOP_TAG 63

typedef __attribute__((ext_vector_type(2))) float v2f;
typedef __attribute__((ext_vector_type(8))) float v8f;
typedef unsigned int v4u __attribute__((ext_vector_type(4)));
typedef unsigned int v8u __attribute__((ext_vector_type(8)));

// Forward-kernel tiling: one TDM transfer covers TCH steps x 16 sequences.
#define TCH 4
#define NCH (TSTEPS / TCH)
#define ROWDW (TCH * 65)  // LDS DWORDs per sequence row after TDM padding (4*64 + 4 pads)

// ---------------------------------------------------------------------------
// Kernel 0: E = exp(trans). Masked entries (-10000) underflow to exactly 0,
// reproducing the logsumexp masking in exp-space.
// ---------------------------------------------------------------------------
__global__ __launch_bounds__(256) void prep_exp_kernel(const float* __restrict__ trans,
                                                       float* __restrict__ E) {
  int i = blockIdx.x * 256 + threadIdx.x;
  if (i < TAGS * TAGS) E[i] = __expf(trans[i]);
}

// ---------------------------------------------------------------------------
// TDM: DMA one [16 seq] x [TCH*64 float] tile into LDS, inserting 1 DWORD of
// padding after every 64 DWORDs (pad_interval=5 -> 64 DW, pad_amount=0 -> 1 DW)
// so each 64-tag step lands at stride 65 (bank-conflict-free B-operand reads).
// D# per CDNA5 ISA; issued via inline asm (portable
// across the 5-arg / 6-arg builtin toolchains). Tracked by TENSORcnt.
// ---------------------------------------------------------------------------
__device__ __forceinline__ void tdm_load_chunk(const float* gbase, unsigned ldsByteAddr) {
  unsigned long long ga = (unsigned long long)(uintptr_t)gbase;
  v4u g0;
  g0[0] = 1u;                                   // count=1 (valid), no gather/restore
  g0[1] = ldsByteAddr;                          // lds_addr (wave-relative byte offset)
  g0[2] = (unsigned)ga;                         // global_addr[31:0]
  g0[3] = (unsigned)((ga >> 32) & 0x1FFFFFFu)   // global_addr[56:32]
          | 0x80000000u;                        // type=2 ("image")
  v8u g1;
  g1[0] = (2u << 16)        // data_size = 4 bytes
        | (1u << 20)        // pad_enable
        | (5u << 22);       // pad_interval: every 64 DWORDs   (pad_amount=0 -> 1 DWORD)
  g1[1] = ((unsigned)(TCH * TAGS) << 16);   // tensor_dim0 = 256 elements (row length)
  g1[2] = (16u << 16);                      // tensor_dim1 = 16 rows
  g1[3] = ((unsigned)(TCH * TAGS) << 16);   // tile_dim0 = 256
  g1[4] = 16u;                              // tile_dim1 = 16, tile_dim2 = 0
  g1[5] = (unsigned)(TSTEPS * TAGS);        // tensor_dim0_stride = T*K elements
  g1[6] = 0u;
  g1[7] = 0u;
  v4u gz = {0u, 0u, 0u, 0u};                // groups 2/3 unused (2D tensor)
  asm volatile("tensor_load_to_lds %0, %1, %2, %3"
               :
               : "s"(g0), "s"(g1), "s"(gz), "s"(gz)
               : "memory");
}

// ---------------------------------------------------------------------------
// Kernel 1: forward algorithm. One wave (32 lanes) per 16 sequences.
// Each step: fv'[next] = feat[next] + m + log( (E @ exp(fv - m))[next] )
// 64x64x16 GEMM per step via V_WMMA_F32_16X16X4_F32 (4 M-tiles x 16 K-chunks),
// A (E tiles) resident in 128 VGPRs, feats streamed by the Tensor Data Mover.
// ---------------------------------------------------------------------------
__global__ __launch_bounds__(32) void crf_forward_kernel(
    const float* __restrict__ feats, const float* __restrict__ trans,
    const float* __restrict__ E, float* __restrict__ alphaOut) {
  const int lane = threadIdx.x;
  const int n = lane & 15;   // sequence column (WMMA N index)
  const int h = lane >> 4;   // lane half
  const int b0 = blockIdx.x * 16;

  __shared__ float fbuf[2][16 * ROWDW];  // TDM-filled feat chunks (double buffered)
  __shared__ float ubuf[16 * 68];        // u = exp(fv-m); stride 68 -> conflict-free b64 reads
  __shared__ float redm[2][16];          // cross-half max reduce
  __shared__ float redsum[2][16];        // cross-half sum reduce (terminal)

  // Preload A operands: E 16x4 tiles. Lanes 0-15 hold K={0,1}, 16-31 K={2,3}.
  v2f a[4][16];
#pragma unroll
  for (int mt = 0; mt < 4; ++mt)
#pragma unroll
    for (int kc = 0; kc < 16; ++kc)
      a[mt][kc] = *(const v2f*)(E + (mt * 16 + n) * TAGS + kc * 4 + 2 * h);

  // init: fv0 = -1e4 except START=0  =>  m=0, u = one-hot(START)
#pragma unroll
  for (int i = 0; i < 32; ++i) {
    int idx = lane + 32 * i;
    int s = idx >> 6, tg = idx & 63;
    ubuf[s * 68 + tg] = (tg == START_TAG) ? 1.0f : 0.0f;
  }
  float m = 0.0f;
  float fv[4][8];

  const float* base0 = feats + (size_t)b0 * TSTEPS * TAGS;
  const unsigned lds0 = (unsigned)(uintptr_t)&fbuf[0][0];
  const unsigned lds1 = (unsigned)(uintptr_t)&fbuf[1][0];

  tdm_load_chunk(base0, lds0);  // prefetch chunk 0

  for (int c = 0; c < NCH; ++c) {
    if (c + 1 < NCH) {  // issue next chunk, then wait only for the current one
      tdm_load_chunk(base0 + (size_t)(c + 1) * TCH * TAGS, ((c + 1) & 1) ? lds1 : lds0);
      __builtin_amdgcn_s_wait_tensorcnt(1);
    } else {
      __builtin_amdgcn_s_wait_tensorcnt(0);
    }
    asm volatile("" ::: "memory");  // keep LDS reads below the wait
    const float* fb = &fbuf[c & 1][0];

#pragma unroll
    for (int tt = 0; tt < TCH; ++tt) {
      // s[tag][seq] = (E @ u)  via 64 f32 WMMAs
      v8f acc[4];
#pragma unroll
      for (int mt = 0; mt < 4; ++mt)
#pragma unroll
        for (int r = 0; r < 8; ++r) acc[mt][r] = 0.0f;

#pragma unroll
      for (int kc = 0; kc < 16; ++kc) {
        v2f bv = *(const v2f*)&ubuf[n * 68 + kc * 4 + 2 * h];  // B: K=kc*4+2h,+1 col n
#pragma unroll
        for (int mt = 0; mt < 4; ++mt)
          acc[mt] = __builtin_amdgcn_wmma_f32_16x16x4_f32(
              false, a[mt][kc], false, bv, (short)0, acc[mt], false, false);
      }

      // fv = log(s) + feat + m ; running per-sequence max
      float mx = -3.4e38f;
#pragma unroll
      for (int mt = 0; mt < 4; ++mt)
#pragma unroll
        for (int r = 0; r < 8; ++r) {
          int tg = mt * 16 + r + 8 * h;  // C/D layout: VGPR r -> M=r (+8 high half)
          float v = __logf(acc[mt][r]) + fb[n * ROWDW + tt * 65 + tg] + m;
          fv[mt][r] = v;
          mx = fmaxf(mx, v);
        }
      redm[h][n] = mx;  // single wave: DS in-order, no barrier needed
      float mnew = fmaxf(mx, redm[h ^ 1][n]);

      // u = exp(fv - mnew) back to LDS for the next step's B operands
#pragma unroll
      for (int mt = 0; mt < 4; ++mt)
#pragma unroll
        for (int r = 0; r < 8; ++r) {
          int tg = mt * 16 + r + 8 * h;
          ubuf[n * 68 + tg] = __expf(fv[mt][r] - mnew);
        }
      m = mnew;
    }
  }

  // terminal: alpha = logsumexp(fv + trans[STOP, :])
  float tv[4][8];
  float mx = -3.4e38f;
#pragma unroll
  for (int mt = 0; mt < 4; ++mt)
#pragma unroll
    for (int r = 0; r < 8; ++r) {
      int tg = mt * 16 + r + 8 * h;
      float v = fv[mt][r] + trans[STOP_TAG * TAGS + tg];
      tv[mt][r] = v;
      mx = fmaxf(mx, v);
    }
  redm[h][n] = mx;
  float M = fmaxf(mx, redm[h ^ 1][n]);
  float s = 0.0f;
#pragma unroll
  for (int mt = 0; mt < 4; ++mt)
#pragma unroll
    for (int r = 0; r < 8; ++r) s += __expf(tv[mt][r] - M);
  redsum[h][n] = s;
  float total = s + redsum[h ^ 1][n];
  if (h == 0) alphaOut[b0 + n] = M + __logf(total);
}

// ---------------------------------------------------------------------------
// Kernel 2: Viterbi (max-plus, VALU) + gold score + nll. One block (64 lanes,
// 2 waves) per sequence; lane = next-tag. Backpointers live entirely in LDS.
// ---------------------------------------------------------------------------
__global__ __launch_bounds__(64) void crf_viterbi_kernel(
    const float* __restrict__ feats, const float* __restrict__ trans,
    const int* __restrict__ tags, const float* __restrict__ alphaWs,
    float* __restrict__ out) {
  const int b = blockIdx.x;
  const int tid = threadIdx.x;

  __shared__ float transT[64 * 65];                 // transT[prev][next] = trans[next][prev]
  __shared__ __align__(16) float fvb[2][64];        // double-buffered viterbi state
  __shared__ unsigned char bp8[TSTEPS * 64];        // backpointers
  __shared__ float red[64];
  __shared__ int pathL[TSTEPS];

  for (int i = 0; i < 64; ++i) transT[tid * 65 + i] = trans[i * 64 + tid];
  fvb[0][tid] = (tid == START_TAG) ? 0.0f : -10000.0f;
  __syncthreads();

  // this lane's transition row trans[next=tid][prev=0..63] into registers
  float tr[64];
#pragma unroll
  for (int p = 0; p < 64; ++p) tr[p] = transT[p * 65 + tid];

  const float* fb = feats + (size_t)b * TSTEPS * TAGS;

  for (int t = 0; t < TSTEPS; ++t) {
    const float* fvp = fvb[t & 1];
    float best = -3.4e38f;
    int bp = 0;
#pragma unroll
    for (int p4 = 0; p4 < 16; ++p4) {  // broadcast fv as float4: 16 DS ops, not 64
      float4 q = *(const float4*)&fvp[p4 * 4];
      float v0 = q.x + tr[4 * p4 + 0];
      if (v0 > best) { best = v0; bp = 4 * p4 + 0; }
      float v1 = q.y + tr[4 * p4 + 1];
      if (v1 > best) { best = v1; bp = 4 * p4 + 1; }
      float v2 = q.z + tr[4 * p4 + 2];
      if (v2 > best) { best = v2; bp = 4 * p4 + 2; }
      float v3 = q.w + tr[4 * p4 + 3];
      if (v3 > best) { best = v3; bp = 4 * p4 + 3; }
    }
    bp8[t * 64 + tid] = (unsigned char)bp;
    fvb[(t + 1) & 1][tid] = best + fb[(size_t)t * TAGS + tid];
    __syncthreads();
  }

  // terminal + argmax over tags
  float term = fvb[TSTEPS & 1][tid] + transT[tid * 65 + STOP_TAG];  // + trans[STOP][tid]
  red[tid] = term;
  __syncthreads();
  if (tid == 0) {
    int last = 0;
    float bs = red[0];
    for (int i = 1; i < 64; ++i)
      if (red[i] > bs) { bs = red[i]; last = i; }
    out[BATCH + b] = bs;  // path_score
    int c = last;
    for (int t = TSTEPS - 1; t >= 0; --t) {  // backtrace from LDS
      pathL[t] = c;
      c = bp8[t * 64 + c];
    }
  }
  __syncthreads();
#pragma unroll
  for (int i = 0; i < TSTEPS / 64; ++i) {
    int t = tid + i * 64;
    out[2 * BATCH + (size_t)b * TSTEPS + t] = (float)pathL[t];  // paths (float out)
  }

  // gold score: sum trans[pad_stop, pad_start] + emissions
  float g = 0.0f;
#pragma unroll
  for (int i = 0; i < TSTEPS / 64; ++i) {
    int t = tid + i * 64;
    int tg = tags[(size_t)b * TSTEPS + t];
    int pv = (t == 0) ? START_TAG : tags[(size_t)b * TSTEPS + t - 1];
    g += transT[pv * 65 + tg] + fb[(size_t)t * TAGS + tg];  // trans[tg][pv] + emit
  }
  if (tid == 0) g += transT[tags[(size_t)b * TSTEPS + TSTEPS - 1] * 65 + STOP_TAG];
  red[tid] = g;
  __syncthreads();
  if (tid == 0) {
    float gold = 0.0f;
    for (int i = 0; i < 64; ++i) gold += red[i];
    out[b] = alphaWs[b] - gold;  // nll
  }
}

// ---------------------------------------------------------------------------
extern "C" void kernel_launch(void* const* d_in, const int* in_sizes, int n_in,
                              void* d_out, int out_size, void* d_ws, size_t ws_size,
                              hipStream_t stream) {
  const float* feats = (const float*)d_in[0];  // [B, T, K] f32
  const float* trans = (const float*)d_in[1];  // [K, K]   f32
  const int* tags = (const int*)d_in[2];       // [B, T]   i32
  float* out = (float*)d_out;                  // [B nll | B score | B*T path]

  float* E = (float*)d_ws;           // 4096 floats
  float* alphaWs = E + TAGS * TAGS;  // 1024 floats

  prep_exp_kernel<<<16, 256, 0, stream>>>(trans, E);
  crf_forward_kernel<<<BATCH / 16, 32, 0, stream>>>(feats, trans, E, alphaWs);
  crf_viterbi_kernel<<<BATCH, 64, 0, stream>>>(feats, trans, tags, alphaWs, out);
}